// RecurrentGCN_10264971838234
// MI455X (gfx1250) — compile-verified
//
#include <hip/hip_runtime.h>
#include <hip/hip_bf16.h>
#include <stdint.h>

typedef __attribute__((ext_vector_type(2))) float v2f;
typedef __attribute__((ext_vector_type(8))) float v8f;

#define NF 64   // node features
#define NC 32   // GRU hidden / filters

// ---------------- zero fill (Tx + deg) ----------------
__global__ void k_fill_zero(float* __restrict__ p, long long n) {
    long long i = (long long)blockIdx.x * blockDim.x + threadIdx.x;
    if (i < n) p[i] = 0.0f;
}

// ---------------- weighted degree: deg[src] += w ----------------
__global__ void k_deg(const long long* __restrict__ ei, const float* __restrict__ w,
                      float* __restrict__ deg, int E) {
    int e = blockIdx.x * blockDim.x + threadIdx.x;
    if (e < E) atomicAdd(deg + (int)ei[e], w[e]);
}

// ---------------- dinv = deg>0 ? rsqrt(deg) : 0 ----------------
__global__ void k_dinv(const float* __restrict__ deg, float* __restrict__ dinv, int N) {
    int i = blockIdx.x * blockDim.x + threadIdx.x;
    if (i < N) {
        float d = deg[i];
        dinv[i] = (d > 0.0f) ? rsqrtf(d) : 0.0f;
    }
}

// ---------------- edge scatter: Tx[dst] += norm * x[src] ----------------
// 16 lanes per edge, float4 per lane (64 features), 4 f32 atomics per lane.
__global__ void k_scatter(const long long* __restrict__ ei, const float* __restrict__ w,
                          const float* __restrict__ dinv, const float* __restrict__ x,
                          float* __restrict__ Tx, int E) {
    long long t = (long long)blockIdx.x * blockDim.x + threadIdx.x;
    int e = (int)(t >> 4);
    if (e >= E) return;
    int j = (int)(t & 15);
    int s = (int)ei[e];
    int d = (int)ei[(long long)E + e];
    float norm = -w[e] * dinv[s] * dinv[d];
    float4 v = ((const float4*)x)[(size_t)s * 16 + j];
    float* o = Tx + (size_t)d * NF + j * 4;
    atomicAdd(o + 0, norm * v.x);
    atomicAdd(o + 1, norm * v.y);
    atomicAdd(o + 2, norm * v.z);
    atomicAdd(o + 3, norm * v.w);
}

// ---------------- build concatenated weights ----------------
// Wcat (128 x 64): rows 0..63 = W*[0] (applied to x), rows 64..127 = W*[1] (applied to Tx)
// cols 0..31 = z gate (Wxz), cols 32..63 = h gate (Wxh)
// bcat[64]: bz = bxz+bhz, bh = bxh+bhh
__global__ void k_wcat(const float* __restrict__ Wxz, const float* __restrict__ Wxh,
                       const float* __restrict__ bxz, const float* __restrict__ bhz,
                       const float* __restrict__ bxh, const float* __restrict__ bhh,
                       float* __restrict__ Wcat, float* __restrict__ bcat) {
    int t = blockIdx.x * blockDim.x + threadIdx.x;
    if (t < 128 * 64) {
        int k = t >> 6;        // 0..127
        int c = t & 63;        // 0..63
        int kk = (k < 64) ? 0 : 1;
        int f  = (k < 64) ? k : (k - 64);
        const float* W = (c < NC) ? Wxz : Wxh;
        int cc = (c < NC) ? c : (c - NC);
        Wcat[t] = W[kk * (NF * NC) + f * NC + cc];
    }
    if (t < 64) {
        bcat[t] = (t < NC) ? (bxz[t] + bhz[t]) : (bxh[t - NC] + bhh[t - NC]);
    }
}

// ---------------- fused WMMA GEMM + GRU epilogue + projection ----------------
// G(N x 64) = [x | Tx](N x 128) @ Wcat(128 x 64) + bcat
// Z = sigmoid(G[:, :32]); Ht = tanh(G[:, 32:]); H = (1-Z)*Ht
// out = relu(H) @ Wl + bl
// One wave computes a 16-row x 64-col tile via V_WMMA_F32_16X16X4_F32 (K in steps of 4).
// LDS holds weights as K-pair float2 elements: sWp[k2*64+c] = {W[2k2][c], W[2k2+1][c]}
// so every B-fragment is a single aligned 64-bit LDS load (no VGPR marshalling).
__global__ __launch_bounds__(256) void k_gemm_gru(
        const float* __restrict__ x, const float* __restrict__ Tx,
        const float* __restrict__ Wcat, const float* __restrict__ bcat,
        const float* __restrict__ Wl, const float* __restrict__ bl,
        float* __restrict__ out, int N) {
    __shared__ v2f sWp[64 * 64];     // 32 KB staged weights, K-paired

    for (int i = threadIdx.x; i < 64 * 64; i += 256) {
        int k2 = i >> 6, c = i & 63;
        v2f t;
        t.x = Wcat[(2 * k2) * 64 + c];
        t.y = Wcat[(2 * k2 + 1) * 64 + c];
        sWp[i] = t;
    }
    __syncthreads();

    const int wave = threadIdx.x >> 5;
    const int lane = threadIdx.x & 31;
    const int half = lane >> 4;      // 0: K pair {0,1}, 1: K pair {2,3}
    const int l16  = lane & 15;      // A: row index; B/C/D: column index

    const int rowBase = blockIdx.x * 128 + wave * 16;
    int arow = rowBase + l16;
    if (arow > N - 1) arow = N - 1;  // clamp; invalid rows never stored

    const float* xr = x  + (size_t)arow * NF;
    const float* tr = Tx + (size_t)arow * NF;

    v8f acc0 = {}, acc1 = {}, acc2 = {}, acc3 = {};

    // K = 0..63 from x
    #pragma unroll 4
    for (int k0 = 0; k0 < 64; k0 += 4) {
        int ka = k0 + half * 2;
        v2f a = *(const v2f*)(xr + ka);
        const v2f* wb = sWp + ((k0 >> 1) + half) * 64 + l16;
        v2f b0 = wb[0], b1 = wb[16], b2 = wb[32], b3 = wb[48];
        acc0 = __builtin_amdgcn_wmma_f32_16x16x4_f32(false, a, false, b0, (short)0, acc0, false, false);
        acc1 = __builtin_amdgcn_wmma_f32_16x16x4_f32(false, a, false, b1, (short)0, acc1, false, false);
        acc2 = __builtin_amdgcn_wmma_f32_16x16x4_f32(false, a, false, b2, (short)0, acc2, false, false);
        acc3 = __builtin_amdgcn_wmma_f32_16x16x4_f32(false, a, false, b3, (short)0, acc3, false, false);
    }
    // K = 64..127 from Tx
    #pragma unroll 4
    for (int k0 = 0; k0 < 64; k0 += 4) {
        int ka = k0 + half * 2;
        v2f a = *(const v2f*)(tr + ka);
        const v2f* wb = sWp + (32 + (k0 >> 1) + half) * 64 + l16;
        v2f b0 = wb[0], b1 = wb[16], b2 = wb[32], b3 = wb[48];
        acc0 = __builtin_amdgcn_wmma_f32_16x16x4_f32(false, a, false, b0, (short)0, acc0, false, false);
        acc1 = __builtin_amdgcn_wmma_f32_16x16x4_f32(false, a, false, b1, (short)0, acc1, false, false);
        acc2 = __builtin_amdgcn_wmma_f32_16x16x4_f32(false, a, false, b2, (short)0, acc2, false, false);
        acc3 = __builtin_amdgcn_wmma_f32_16x16x4_f32(false, a, false, b3, (short)0, acc3, false, false);
    }

    // Epilogue. Columns: tile0 -> z cols l16, tile1 -> z cols 16+l16,
    //                    tile2 -> h cols l16, tile3 -> h cols 16+l16.
    // z/h pairs share lane and VGPR slot.
    const float b0 = bcat[l16];
    const float b1 = bcat[16 + l16];
    const float b2 = bcat[32 + l16];
    const float b3 = bcat[48 + l16];
    const float wl_a = Wl[l16];
    const float wl_b = Wl[16 + l16];
    const float blv = bl[0];

    #pragma unroll
    for (int v = 0; v < 8; ++v) {
        float z0 = 1.0f / (1.0f + __expf(-(acc0[v] + b0)));
        float h0 = tanhf(acc2[v] + b2);
        float z1 = 1.0f / (1.0f + __expf(-(acc1[v] + b1)));
        float h1 = tanhf(acc3[v] + b3);
        float H0 = (1.0f - z0) * h0;
        float H1 = (1.0f - z1) * h1;
        float p = fmaxf(H0, 0.0f) * wl_a + fmaxf(H1, 0.0f) * wl_b;
        // Reduce over the 16 columns handled by this half-wave
        p += __shfl_xor(p, 1, 32);
        p += __shfl_xor(p, 2, 32);
        p += __shfl_xor(p, 4, 32);
        p += __shfl_xor(p, 8, 32);
        if (l16 == 0) {
            int row = rowBase + half * 8 + v;   // C/D layout: VGPR v -> M = v + 8*half
            if (row < N) out[row] = p + blv;
        }
    }
}

extern "C" void kernel_launch(void* const* d_in, const int* in_sizes, int n_in,
                              void* d_out, int out_size, void* d_ws, size_t ws_size,
                              hipStream_t stream) {
    const float*     x   = (const float*)d_in[0];
    const long long* ei  = (const long long*)d_in[1];   // int64 edge_index (2, E)
    const float*     w   = (const float*)d_in[2];
    const float*     Wxz = (const float*)d_in[3];
    const float*     bxz = (const float*)d_in[4];
    const float*     bhz = (const float*)d_in[6];
    const float*     Wxh = (const float*)d_in[11];
    const float*     bxh = (const float*)d_in[12];
    const float*     bhh = (const float*)d_in[14];
    const float*     Wl  = (const float*)d_in[15];
    const float*     bl  = (const float*)d_in[16];
    float* out = (float*)d_out;

    const int N = in_sizes[0] / NF;
    const int E = in_sizes[2];

    // Workspace layout (floats)
    float* Tx   = (float*)d_ws;                 // N*64
    float* deg  = Tx + (size_t)N * NF;          // N
    float* dinv = deg + N;                      // N
    float* Wcat = dinv + N;                     // 128*64
    float* bcat = Wcat + 128 * 64;              // 64

    const int BT = 256;

    // 1) zero Tx + deg (contiguous)
    long long nz = (long long)N * NF + N;
    k_fill_zero<<<(int)((nz + BT - 1) / BT), BT, 0, stream>>>(Tx, nz);

    // 2) weighted degree
    k_deg<<<(E + BT - 1) / BT, BT, 0, stream>>>(ei, w, deg, E);

    // 3) dinv
    k_dinv<<<(N + BT - 1) / BT, BT, 0, stream>>>(deg, dinv, N);

    // 4) edge scatter (16 lanes per edge)
    long long st = (long long)E * 16;
    k_scatter<<<(int)((st + BT - 1) / BT), BT, 0, stream>>>(ei, w, dinv, x, Tx, E);

    // 5) concatenated weights + biases
    k_wcat<<<(128 * 64 + BT - 1) / BT, BT, 0, stream>>>(Wxz, Wxh, bxz, bhz, bxh, bhh, Wcat, bcat);

    // 6) fused WMMA GEMM + GRU + projection
    int gblocks = (N + 127) / 128;
    k_gemm_gru<<<gblocks, 256, 0, stream>>>(x, Tx, Wcat, bcat, Wl, bl, out, N);
}